// SpikeLayer_66967130079991
// MI455X (gfx1250) — compile-verified
//
#include <hip/hip_runtime.h>

// LIF neuron timestep update (SNN-toolbox SpikeLayer.update_neurons).
// Purely elementwise & streaming: 5 f32 inputs read once, 6 f32 outputs
// written once, no reuse -> bandwidth-bound (~565 MB @ 23.3 TB/s ~ 24 us).
// Strategy: float4 (B128) fully-coalesced accesses with non-temporal cache
// hints so the 565 MB stream doesn't thrash the 192 MB L2.

typedef float v4f __attribute__((ext_vector_type(4)));

#define V_THRESH   1.0f
#define SIM_TIME   0.5f
#define REFRAC_SET 2.5f   // TIME + TAU_REFRAC = 0.5 + 2.0

__global__ __launch_bounds__(256) void lif_update_vec4(
    const v4f* __restrict__ impulse,
    const v4f* __restrict__ mem,
    const v4f* __restrict__ mem_acc,
    const v4f* __restrict__ refrac,
    const v4f* __restrict__ counts,
    float* __restrict__ out,     // 6 stacked tensors of n elements each
    long nvec,                   // n / 4
    long n)                      // elements per tensor
{
    long i = (long)blockIdx.x * blockDim.x + threadIdx.x;
    long stride = (long)gridDim.x * blockDim.x;

    v4f* __restrict__ o_spk  = (v4f*)(out);            // spikes
    v4f* __restrict__ o_mem  = (v4f*)(out + n);        // mem_out
    v4f* __restrict__ o_acc  = (v4f*)(out + 2 * n);    // new_mem_acc
    v4f* __restrict__ o_ref  = (v4f*)(out + 3 * n);    // refrac_out
    v4f* __restrict__ o_cnt  = (v4f*)(out + 4 * n);    // counts_out
    v4f* __restrict__ o_trn  = (v4f*)(out + 5 * n);    // spiketrain

    for (; i < nvec; i += stride) {
        // Streaming loads: NT hint -> global_load_b128 th:NT, bypass-friendly.
        v4f imp  = __builtin_nontemporal_load(&impulse[i]);
        v4f m    = __builtin_nontemporal_load(&mem[i]);
        v4f acc  = __builtin_nontemporal_load(&mem_acc[i]);
        v4f r    = __builtin_nontemporal_load(&refrac[i]);
        v4f c    = __builtin_nontemporal_load(&counts[i]);

        v4f spk, mo, ro, co, st;
#pragma unroll
        for (int k = 0; k < 4; ++k) {
            // Refractory neurons see zero input.
            float mi   = (r[k] > SIM_TIME) ? 0.0f : imp[k];
            float nm   = m[k]   + mi;
            float nacc = acc[k] + mi;
            bool fire  = (nm >= V_THRESH);
            // linear activation: spikes in {0, V_THRESH}; the reference's
            // (spikes < 0) branch is statically dead for V_THRESH > 0.
            spk[k] = fire ? V_THRESH : 0.0f;
            mo[k]  = fire ? (nm - V_THRESH) : nm;      // reset-by-subtraction
            ro[k]  = fire ? REFRAC_SET : r[k];
            co[k]  = c[k] + (fire ? 1.0f : 0.0f);
            st[k]  = fire ? (V_THRESH * SIM_TIME) : 0.0f;
            acc[k] = nacc;
        }

        // Streaming stores: NT hint -> global_store_b128 th:NT (write-rinse).
        __builtin_nontemporal_store(spk, &o_spk[i]);
        __builtin_nontemporal_store(mo,  &o_mem[i]);
        __builtin_nontemporal_store(acc, &o_acc[i]);
        __builtin_nontemporal_store(ro,  &o_ref[i]);
        __builtin_nontemporal_store(co,  &o_cnt[i]);
        __builtin_nontemporal_store(st,  &o_trn[i]);
    }

    // Scalar tail (n not divisible by 4): first few threads of the grid.
    long tail = n - nvec * 4;
    long tid  = (long)blockIdx.x * blockDim.x + threadIdx.x;
    if (tid < tail) {
        long e = nvec * 4 + tid;
        const float* fi = (const float*)impulse;
        const float* fm = (const float*)mem;
        const float* fa = (const float*)mem_acc;
        const float* fr = (const float*)refrac;
        const float* fc = (const float*)counts;
        float r0 = fr[e];
        float mi = (r0 > SIM_TIME) ? 0.0f : fi[e];
        float nm = fm[e] + mi;
        bool fire = (nm >= V_THRESH);
        out[e]         = fire ? V_THRESH : 0.0f;
        out[n + e]     = fire ? (nm - V_THRESH) : nm;
        out[2 * n + e] = fa[e] + mi;
        out[3 * n + e] = fire ? REFRAC_SET : r0;
        out[4 * n + e] = fc[e] + (fire ? 1.0f : 0.0f);
        out[5 * n + e] = fire ? (V_THRESH * SIM_TIME) : 0.0f;
    }
}

extern "C" void kernel_launch(void* const* d_in, const int* in_sizes, int n_in,
                              void* d_out, int out_size, void* d_ws, size_t ws_size,
                              hipStream_t stream) {
    (void)n_in; (void)out_size; (void)d_ws; (void)ws_size;

    const v4f* impulse = (const v4f*)d_in[0];
    const v4f* mem     = (const v4f*)d_in[1];
    const v4f* mem_acc = (const v4f*)d_in[2];
    const v4f* refrac  = (const v4f*)d_in[3];
    const v4f* counts  = (const v4f*)d_in[4];
    float* out = (float*)d_out;

    long n    = (long)in_sizes[0];   // 12,845,056 elements per tensor
    long nvec = n / 4;               // float4 groups

    const int block = 256;           // 8 wave32 waves per block
    long blocks = (nvec + block - 1) / block;
    if (blocks > 1048576) blocks = 1048576;  // grid-stride loop covers rest

    lif_update_vec4<<<(int)blocks, block, 0, stream>>>(
        impulse, mem, mem_acc, refrac, counts, out, nvec, n);
}